// GATLayer_40767829574576
// MI455X (gfx1250) — compile-verified
//
#include <hip/hip_runtime.h>
#include <hip/hip_bf16.h>

typedef __attribute__((ext_vector_type(2))) float        v2f;
typedef __attribute__((ext_vector_type(8))) float        v8f;
typedef __attribute__((ext_vector_type(4))) unsigned int v4u;
typedef __attribute__((ext_vector_type(8))) int          v8i;
typedef __attribute__((ext_vector_type(4))) int          v4i;

#define IN_F 256
#define OUT_F 64

// ---------------------------------------------------------------------------
// Kernel 1: hp[N,64] = h[N,256] @ W[256,64] with V_WMMA_F32_16X16X4_F32.
// W (64 KB) is staged once per workgroup into LDS by the Tensor Data Mover
// (tensor_load_to_lds, TENSORcnt), then every wave feeds B-frags from LDS.
// One wave per 16-row tile; all four 16-col output tiles kept in registers.
// ---------------------------------------------------------------------------
__global__ void gat_gemm_wmma(const float* __restrict__ h,
                              const float* __restrict__ W,
                              float* __restrict__ hp,
                              int nRowTiles) {
  __shared__ float lds_w[IN_F * OUT_F];   // 64 KB, offset 0 in LDS

  // --- TDM: DMA W[256,64] f32 into LDS (wave 0 issues; EXEC-independent) ---
  if (threadIdx.x < 32) {
    const unsigned long long gaddr = (unsigned long long)(const void*)W;
    v4u g0;
    g0.x = 1u;                                   // count=1, user descriptor
    g0.y = 0u;                                   // lds_addr = 0 (lds_w base)
    g0.z = (unsigned)(gaddr & 0xFFFFFFFFull);    // global_addr[31:0]
    g0.w = (unsigned)((gaddr >> 32) & 0x01FFFFFFu) | (2u << 30); // addr[56:32] | type=2

    v8i g1;
    g1[0] = (int)(2u << 16);                     // data_size=2 (4 bytes)
    g1[1] = (int)(OUT_F << 16);                  // tensor_dim0 = 64 (bits 79:48 lo)
    g1[2] = (int)(IN_F << 16);                   // tensor_dim1 = 256 (bits 111:80 lo)
    g1[3] = (int)(OUT_F << 16);                  // tile_dim0 = 64 (bits 127:112)
    g1[4] = (int)IN_F;                           // tile_dim1 = 256 (bits 143:128)
    g1[5] = (int)OUT_F;                          // tensor_dim0_stride = 64
    g1[6] = 0;
    g1[7] = 0;

    v4i gz4 = {0, 0, 0, 0};                      // groups 2/3 unused (2-D tensor)
    v8i gz8 = {0, 0, 0, 0, 0, 0, 0, 0};
    __builtin_amdgcn_tensor_load_to_lds(g0, g1, gz4, gz4, gz8, 0);
    __builtin_amdgcn_s_wait_tensorcnt(0);
  }

  // The TDM builtin's LDS write is via a descriptor field, invisible to alias
  // analysis; without this, LLVM folds the never-"stored" lds_w loads to undef.
  // Escape the pointer through an asm with a memory clobber so the compiler
  // must assume lds_w now holds the DMA'd data.
  {
    void* p = (void*)lds_w;
    asm volatile("" : "+v"(p) :: "memory");
  }
  __syncthreads();

  const int wave = (int)((blockIdx.x * blockDim.x + threadIdx.x) >> 5);
  if (wave < nRowTiles) {                        // wave-uniform: EXEC all-1s
    const int lane = threadIdx.x & 31;
    const int half = lane >> 4;                  // 0: K pair {0,1}, 1: {2,3}
    const int l16  = lane & 15;

    const float* __restrict__ arow =
        h + (size_t)(wave * 16 + l16) * IN_F + 2 * half;

    v8f acc0 = {}, acc1 = {}, acc2 = {}, acc3 = {};

    for (int k = 0; k < IN_F; k += 4) {
      v2f a = *(const v2f*)(arow + k);

      const float* __restrict__ wl = lds_w + (k + 2 * half) * OUT_F;
      v2f b0, b1, b2, b3;
      b0.x = wl[l16];       b0.y = wl[OUT_F + l16];
      b1.x = wl[16 + l16];  b1.y = wl[OUT_F + 16 + l16];
      b2.x = wl[32 + l16];  b2.y = wl[OUT_F + 32 + l16];
      b3.x = wl[48 + l16];  b3.y = wl[OUT_F + 48 + l16];

      acc0 = __builtin_amdgcn_wmma_f32_16x16x4_f32(false, a, false, b0, (short)0, acc0, false, false);
      acc1 = __builtin_amdgcn_wmma_f32_16x16x4_f32(false, a, false, b1, (short)0, acc1, false, false);
      acc2 = __builtin_amdgcn_wmma_f32_16x16x4_f32(false, a, false, b2, (short)0, acc2, false, false);
      acc3 = __builtin_amdgcn_wmma_f32_16x16x4_f32(false, a, false, b3, (short)0, acc3, false, false);
    }

    // Store D: vgpr r holds M = r + 8*half, N = l16 (+16/32/48 per col tile)
    for (int r = 0; r < 8; ++r) {
      const int orow = wave * 16 + r + 8 * half;
      float* __restrict__ o = hp + (size_t)orow * OUT_F + l16;
      o[0]  = acc0[r];
      o[16] = acc1[r];
      o[32] = acc2[r];
      o[48] = acc3[r];
    }
  }
}

// ---------------------------------------------------------------------------
// Kernel 2: zero d_out (accumulator) and deg; reduce attention_w -> s_aw.
// ---------------------------------------------------------------------------
__global__ void gat_init(float* __restrict__ out, float* __restrict__ deg,
                         const float* __restrict__ aw, float* __restrict__ s_aw,
                         int heads, int total, int n) {
  const int tid    = (int)(blockIdx.x * blockDim.x + threadIdx.x);
  const int stride = (int)(gridDim.x * blockDim.x);
  if (tid == 0) {
    float s = 0.f;
    for (int j = 0; j < heads; ++j) s += aw[j];
    *s_aw = s;
  }
  for (int i = tid; i < total; i += stride) out[i] = 0.f;
  for (int i = tid; i < n; i += stride) deg[i] = 0.f;
}

// ---------------------------------------------------------------------------
// Kernel 3: one wave per edge; lane owns 2 of the 64 features.
// dot via shuffle butterfly; hardware f32 atomics into L2-resident out.
// ---------------------------------------------------------------------------
__global__ void gat_edge(const float* __restrict__ hp,
                         const int* __restrict__ src,
                         const int* __restrict__ dst,
                         const float* __restrict__ s_aw_ptr,
                         float* __restrict__ out,
                         float* __restrict__ deg,
                         int E) {
  const int e = (int)((blockIdx.x * blockDim.x + threadIdx.x) >> 5);
  if (e >= E) return;                            // wave-uniform
  const int lane = threadIdx.x & 31;

  const int s = src[e];
  const int d = dst[e];

  v2f hs = *(const v2f*)(hp + (size_t)s * OUT_F + lane * 2);
  v2f hd = *(const v2f*)(hp + (size_t)d * OUT_F + lane * 2);

  float p = hs.x * hd.x + hs.y * hd.y;
  for (int off = 16; off > 0; off >>= 1) p += __shfl_xor(p, off, 32);

  const float coef = p * (*s_aw_ptr);

  float* __restrict__ o = out + (size_t)d * OUT_F + lane * 2;
  unsafeAtomicAdd(o,     coef * hs.x);
  unsafeAtomicAdd(o + 1, coef * hs.y);
  if (lane == 0) unsafeAtomicAdd(deg + d, 1.0f);
}

// ---------------------------------------------------------------------------
// Kernel 4: zero in-degree nodes keep hp.
// ---------------------------------------------------------------------------
__global__ void gat_finalize(const float* __restrict__ hp,
                             const float* __restrict__ deg,
                             float* __restrict__ out,
                             int total) {
  const int i = (int)(blockIdx.x * blockDim.x + threadIdx.x);
  if (i >= total) return;
  const int node = i >> 6;   // /OUT_F
  if (deg[node] == 0.f) out[i] = hp[i];
}

extern "C" void kernel_launch(void* const* d_in, const int* in_sizes, int n_in,
                              void* d_out, int out_size, void* d_ws, size_t ws_size,
                              hipStream_t stream) {
  const float* h  = (const float*)d_in[0];   // [N, 256]
  const float* W  = (const float*)d_in[1];   // [256, 64]
  const float* aw = (const float*)d_in[2];   // [1, HEADS]
  const int* src  = (const int*)d_in[3];     // [E]
  const int* dst  = (const int*)d_in[4];     // [E]
  float* out = (float*)d_out;                // [N, 64]

  const int N     = in_sizes[0] / IN_F;
  const int heads = in_sizes[2];
  const int E     = in_sizes[3];
  const int total = N * OUT_F;

  // workspace layout: hp | deg | s_aw
  float* hp   = (float*)d_ws;
  float* deg  = hp + (size_t)total;
  float* s_aw = deg + (size_t)N;

  // 1) GEMM via WMMA + TDM-staged B (N = 50000 = 3125 * 16 row tiles)
  const int nRowTiles = (N + 15) / 16;
  {
    const int waves_per_block = 8;             // 256 threads
    const int blocks = (nRowTiles + waves_per_block - 1) / waves_per_block;
    gat_gemm_wmma<<<blocks, 256, 0, stream>>>(h, W, hp, nRowTiles);
  }

  // 2) init accumulators + attention_w reduction
  gat_init<<<1024, 256, 0, stream>>>(out, deg, aw, s_aw, heads, total, N);

  // 3) edge phase: one wave per edge
  {
    const int waves_per_block = 8;             // 256 threads
    const int blocks = (E + waves_per_block - 1) / waves_per_block;
    gat_edge<<<blocks, 256, 0, stream>>>(hp, src, dst, s_aw, out, deg, E);
  }

  // 4) zero-degree fallback
  gat_finalize<<<(total + 255) / 256, 256, 0, stream>>>(hp, deg, out, total);
}